// cNODE_condensed_56547539419172
// MI455X (gfx1250) — compile-verified
//
#include <hip/hip_runtime.h>
#include <stdint.h>

// Reference semantics: y[b,d] = x[b,d] if (# nonzeros of row b at positions < d) < 8192, else 0.
// Embedding lookup and t are dead code w.r.t. the output -> never touched.
// Memory-bound: 128 MB total traffic -> ~5.5 us floor at 23.3 TB/s (fits in 192 MB L2 for replays).

#define N_ROWS   256
#define N_COLS   65536
#define LEN_CAP  8192
#define THREADS  1024                 // 32 wave32 waves per workgroup
#define VEC      4                    // float4 per thread per tile
#define CHUNK    (THREADS * VEC)      // 4096 floats = 16 KB tile
#define ITERS    (N_COLS / CHUNK)     // 16 tiles per row

__global__ __launch_bounds__(THREADS, 1)
void cnode_compact_copy_kernel(const float* __restrict__ x, float* __restrict__ y) {
  // Double-buffered async staging tiles (2 x 16 KB) + double-buffered wave totals.
  __shared__ __align__(16) float stage[2][CHUNK];
  __shared__ unsigned wtot[2][32];

  const int row  = blockIdx.x;
  const int tid  = threadIdx.x;
  const int lane = tid & 31;
  const int wave = tid >> 5;
  const unsigned lane_lt = (1u << lane) - 1u;

  const uint64_t xrow = (uint64_t)(uintptr_t)(x + (size_t)row * N_COLS);
  float* yrow = y + (size_t)row * N_COLS;

  // 32-bit LDS byte offsets of this thread's 16-byte slot in each buffer
  // (low 32 bits of the flat address of an LDS object == wave-relative LDS offset).
  const unsigned lds0 = (unsigned)(uintptr_t)&stage[0][tid * VEC];
  const unsigned lds1 = (unsigned)(uintptr_t)&stage[1][tid * VEC];

  // Kick off tile 0: HBM -> LDS, tracked by ASYNCcnt, no VGPR round-trip.
  {
    unsigned goff = (unsigned)(tid * VEC * sizeof(float));
    asm volatile("global_load_async_to_lds_b128 %0, %1, %2"
                 :: "v"(lds0), "v"(goff), "s"(xrow) : "memory");
  }

  unsigned row_base = 0;  // running nonzero count of this row (identical in all threads)

  for (int it = 0; it < ITERS; ++it) {
    if (it + 1 < ITERS) {
      // Prefetch next tile into the other buffer, then retire the current one.
      unsigned goff = (unsigned)(((it + 1) * CHUNK + tid * VEC) * sizeof(float));
      unsigned dst  = (it & 1) ? lds0 : lds1;
      asm volatile("global_load_async_to_lds_b128 %0, %1, %2"
                   :: "v"(dst), "v"(goff), "s"(xrow) : "memory");
      asm volatile("s_wait_asynccnt 1" ::: "memory");  // in-order retire -> tile 'it' is in LDS
    } else {
      asm volatile("s_wait_asynccnt 0" ::: "memory");
    }

    // Each wave reads back only the 512 B it requested -> no barrier needed for the staging data.
    const float4 v = *reinterpret_cast<const float4*>(&stage[it & 1][tid * VEC]);

    // Intra-wave nonzero ranks via wave32 ballots (element order = lane*4 + slot).
    unsigned b0 = __builtin_amdgcn_ballot_w32(v.x != 0.0f);
    unsigned b1 = __builtin_amdgcn_ballot_w32(v.y != 0.0f);
    unsigned b2 = __builtin_amdgcn_ballot_w32(v.z != 0.0f);
    unsigned b3 = __builtin_amdgcn_ballot_w32(v.w != 0.0f);

    unsigned below = __popc(b0 & lane_lt) + __popc(b1 & lane_lt) +
                     __popc(b2 & lane_lt) + __popc(b3 & lane_lt);
    unsigned wsum  = __popc(b0) + __popc(b1) + __popc(b2) + __popc(b3);

    // Cross-wave exclusive prefix of the 32 wave totals (double-buffered: 1 barrier/iter).
    const int p = it & 1;
    if (lane == 0) wtot[p][wave] = wsum;
    __syncthreads();

    unsigned wt  = wtot[p][lane];                 // lane l holds wave l's total
    unsigned pre = (lane < wave) ? wt : 0u;
#pragma unroll
    for (int s = 16; s >= 1; s >>= 1) {
      pre += __shfl_xor(pre, s, 32);              // sum over waves < my wave (uniform in wave)
      wt  += __shfl_xor(wt,  s, 32);              // block total (uniform everywhere)
    }

    // Rank of each of my 4 elements among this row's nonzeros (ascending order).
    unsigned r = row_base + pre + below;
    float4 o;
    o.x = (r < LEN_CAP) ? v.x : 0.0f; r += (v.x != 0.0f) ? 1u : 0u;
    o.y = (r < LEN_CAP) ? v.y : 0.0f; r += (v.y != 0.0f) ? 1u : 0u;
    o.z = (r < LEN_CAP) ? v.z : 0.0f; r += (v.z != 0.0f) ? 1u : 0u;
    o.w = (r < LEN_CAP) ? v.w : 0.0f;

    *reinterpret_cast<float4*>(&yrow[it * CHUNK + tid * VEC]) = o;  // global_store_b128

    row_base += wt;  // same value added by every thread -> stays uniform, no broadcast
  }
}

extern "C" void kernel_launch(void* const* d_in, const int* in_sizes, int n_in,
                              void* d_out, int out_size, void* d_ws, size_t ws_size,
                              hipStream_t stream) {
  // setup_inputs order: d_in[0]=t (unused), d_in[1]=x (256*65536 f32), d_in[2]=embed_table (dead).
  const float* x = (const float*)d_in[1];
  float* y = (float*)d_out;
  (void)in_sizes; (void)n_in; (void)d_ws; (void)ws_size; (void)out_size;

  hipLaunchKernelGGL(cnode_compact_copy_kernel, dim3(N_ROWS), dim3(THREADS), 0, stream, x, y);
}